// spatial_attention_34162169872570
// MI455X (gfx1250) — compile-verified
//
#include <hip/hip_runtime.h>
#include <hip/hip_bf16.h>
#include <math.h>

#define MDIM 4096
#define FDIM 128
#define EDIM 128

typedef float v2f_t __attribute__((ext_vector_type(2)));
typedef float v4f_t __attribute__((ext_vector_type(4)));
typedef float v8f_t __attribute__((ext_vector_type(8)));

// ---------------------------------------------------------------------------
// Kernel 0: zero the colsum accumulators (d_ws is poisoned by the harness)
// ---------------------------------------------------------------------------
__global__ void k_zero(float* __restrict__ p, int n) {
    int i = blockIdx.x * blockDim.x + threadIdx.x;
    if (i < n) p[i] = 0.0f;
}

// ---------------------------------------------------------------------------
// Kernel 1: w1 = W_att @ a1, w2 = W_att @ a2   (F=128 rows, 1 block)
// ---------------------------------------------------------------------------
__global__ void k_proj_a(const float* __restrict__ W_att,
                         const float* __restrict__ a1,
                         const float* __restrict__ a2,
                         float* __restrict__ w1, float* __restrict__ w2) {
    int f = threadIdx.x;  // 0..127
    float s1 = 0.f, s2 = 0.f;
    for (int e = 0; e < EDIM; ++e) {
        float w = W_att[f * EDIM + e];
        s1 += w * a1[e];
        s2 += w * a2[e];
    }
    w1[f] = s1;
    w2[f] = s2;
}

// ---------------------------------------------------------------------------
// Kernel 2: s[i] = t[i,:]·w1 ; u[i] = t[i,:]·w2   (b128 row reads)
// ---------------------------------------------------------------------------
__global__ void k_su(const float* __restrict__ t,
                     const float* __restrict__ w1, const float* __restrict__ w2,
                     float* __restrict__ s, float* __restrict__ u) {
    int i = blockIdx.x * blockDim.x + threadIdx.x;
    if (i >= MDIM) return;
    const v4f_t* row4 = (const v4f_t*)(t + (size_t)i * FDIM);
    const v4f_t* w14  = (const v4f_t*)w1;
    const v4f_t* w24  = (const v4f_t*)w2;
    float s1 = 0.f, s2 = 0.f;
    for (int f4 = 0; f4 < FDIM / 4; ++f4) {
        v4f_t v = row4[f4];
        v4f_t a = w14[f4];
        v4f_t b = w24[f4];
        s1 += v.x * a.x + v.y * a.y + v.z * a.z + v.w * a.w;
        s2 += v.x * b.x + v.y * b.y + v.z * b.z + v.w * b.w;
    }
    s[i] = s1;
    u[i] = s2;
}

// ---------------------------------------------------------------------------
// Kernel 3: r[i] = (sum_j nbr[i,j]*u[j]) / max(sum_j nbr[i,j], 1)
// gridDim.y = 3 directions; b128 non-temporal loads (stream-once data).
// One 256-thread block per row: 1024 float4 per row, 4 per thread.
// ---------------------------------------------------------------------------
__global__ __launch_bounds__(256) void k_rowavg(const float* __restrict__ nbr_f,
                                                const float* __restrict__ nbr_b,
                                                const float* __restrict__ nbr_g,
                                                const float* __restrict__ u,
                                                float* __restrict__ r_f,
                                                float* __restrict__ r_b,
                                                float* __restrict__ r_g) {
    const float* nbr = (blockIdx.y == 0) ? nbr_f : (blockIdx.y == 1) ? nbr_b : nbr_g;
    float* r         = (blockIdx.y == 0) ? r_f   : (blockIdx.y == 1) ? r_b   : r_g;
    int i = blockIdx.x;
    int tid = threadIdx.x;
    const v4f_t* row4 = (const v4f_t*)(nbr + (size_t)i * MDIM);
    const v4f_t* u4   = (const v4f_t*)u;
    float acc = 0.f, deg = 0.f;
    for (int j4 = tid; j4 < MDIM / 4; j4 += 256) {
        v4f_t a  = __builtin_nontemporal_load(&row4[j4]);  // stream-once: bypass cache retention
        v4f_t uu = u4[j4];
        acc += a.x * uu.x + a.y * uu.y + a.z * uu.z + a.w * uu.w;
        deg += a.x + a.y + a.z + a.w;
    }
    __shared__ float sa[256];
    __shared__ float sd[256];
    sa[tid] = acc; sd[tid] = deg;
    __syncthreads();
    for (int off = 128; off > 0; off >>= 1) {
        if (tid < off) { sa[tid] += sa[tid + off]; sd[tid] += sd[tid + off]; }
        __syncthreads();
    }
    if (tid == 0) r[i] = sa[0] / fmaxf(sd[0], 1.0f);
}

// ---------------------------------------------------------------------------
// Kernel 4: masked-softmax column sums, gridDim.y = 3 directions.
// e_ij = leakyrelu(s_i + r_j) where adj>0 else -inf (reference's -9e15:
// exp underflows to exactly 0 in f32 either way). Phase A: online (max,sum)
// per row with b128 loads (default RT policy -> fills L2, 64MB << 192MB L2).
// Phase B: re-read row (L2 hit) and scatter att into c via atomics (~2%
// density -> ~82 atomics per 4096-element row).
// ---------------------------------------------------------------------------
__global__ __launch_bounds__(256) void k_att(const float* __restrict__ adj_f,
                                             const float* __restrict__ adj_b,
                                             const float* __restrict__ adj_g,
                                             const float* __restrict__ s,
                                             const float* __restrict__ r_f,
                                             const float* __restrict__ r_b,
                                             const float* __restrict__ r_g,
                                             float* __restrict__ c_f,
                                             float* __restrict__ c_b,
                                             float* __restrict__ c_g) {
    const float NEG_INF = -__builtin_inff();
    const float* adj = (blockIdx.y == 0) ? adj_f : (blockIdx.y == 1) ? adj_b : adj_g;
    const float* r   = (blockIdx.y == 0) ? r_f   : (blockIdx.y == 1) ? r_b   : r_g;
    float* c         = (blockIdx.y == 0) ? c_f   : (blockIdx.y == 1) ? c_b   : c_g;
    int i = blockIdx.x;
    int tid = threadIdx.x;
    const v4f_t* row4 = (const v4f_t*)(adj + (size_t)i * MDIM);
    const v4f_t* r4   = (const v4f_t*)r;
    float si = s[i];

    // phase A: online softmax statistics
    float mx = NEG_INF, sm = 0.f;
    for (int j4 = tid; j4 < MDIM / 4; j4 += 256) {
        v4f_t a  = row4[j4];
        v4f_t rr = r4[j4];
#define ONLINE(av, rv)                                   \
        if ((av) > 0.f) {                                \
            float e = si + (rv);                         \
            e = (e > 0.f) ? e : 0.2f * e;                \
            if (e > mx) {                                \
                sm = sm * expf(mx - e) + 1.0f;           \
                mx = e;                                  \
            } else {                                     \
                sm += expf(e - mx);                      \
            }                                            \
        }
        ONLINE(a.x, rr.x) ONLINE(a.y, rr.y) ONLINE(a.z, rr.z) ONLINE(a.w, rr.w)
#undef ONLINE
    }
    __shared__ float smx[256];
    __shared__ float ssm[256];
    smx[tid] = mx; ssm[tid] = sm;
    __syncthreads();
    for (int off = 128; off > 0; off >>= 1) {
        if (tid < off) {
            float m1 = smx[tid], m2 = smx[tid + off];
            float M = fmaxf(m1, m2);
            float S = 0.f;
            if (m1 != NEG_INF) S += ssm[tid] * expf(m1 - M);
            if (m2 != NEG_INF) S += ssm[tid + off] * expf(m2 - M);
            smx[tid] = M; ssm[tid] = S;
        }
        __syncthreads();
    }
    mx = smx[0];
    sm = ssm[0];
    float inv = (sm > 0.f) ? 1.0f / sm : 0.f;

    // phase B: scatter normalized attention into column sums (row hits L2)
    for (int j4 = tid; j4 < MDIM / 4; j4 += 256) {
        v4f_t a  = row4[j4];
        v4f_t rr = r4[j4];
        int j = j4 * 4;
#define SCATTER(av, rv, jj)                              \
        if ((av) > 0.f) {                                \
            float e = si + (rv);                         \
            e = (e > 0.f) ? e : 0.2f * e;                \
            atomicAdd(&c[jj], expf(e - mx) * inv);       \
        }
        SCATTER(a.x, rr.x, j) SCATTER(a.y, rr.y, j + 1)
        SCATTER(a.z, rr.z, j + 2) SCATTER(a.w, rr.w, j + 3)
#undef SCATTER
    }
}

// ---------------------------------------------------------------------------
// Kernel 5: output GEMMs via V_WMMA_F32_16X16X4_F32 (exact f32 path).
//   y = weight @ t_flat   (t_flat[f,m] = tbuf[f*4096+m], raw reshape)
//   x = weight @ t^T      (x[e,m] = sum_f weight[e,f]*t[m,f])
//   out[m, 0:128]=y*c_f ; out[m,128:256]=x ; out[m,256:384]=y*c_g ; out[m,384:512]=y*c_b
// One wave per 16(e)x16(m) tile; A (weight tile) shared by both WMMA chains.
// ISA layouts: A lane L -> row L%16, K={0,1}|{2,3} by lane half (v0=K even, v1=K odd);
//              B lane L -> col L%16, same K split; D lane L -> col L%16, row v+8*(L/16).
// ---------------------------------------------------------------------------
__global__ __launch_bounds__(256) void k_out(const float* __restrict__ t,
                                             const float* __restrict__ weight,
                                             const float* __restrict__ cf,
                                             const float* __restrict__ cg,
                                             const float* __restrict__ cb,
                                             float* __restrict__ out) {
    int lane  = threadIdx.x & 31;
    int wave  = threadIdx.x >> 5;   // 0..7 -> e-tile
    int mtile = blockIdx.x;         // 0..255 -> m-tile
    int n    = lane & 15;           // N index within tile
    int half = lane >> 4;           // lane half selects K pair
    int m0   = mtile * 16;
    int erow = wave * 16 + n;       // A-matrix row for this lane

    v8f_t dy = {};
    v8f_t dx = {};
    for (int k = 0; k < FDIM; k += 4) {
        int ka = k + half * 2;
        v2f_t A;
        A.x = weight[erow * FDIM + ka];
        A.y = weight[erow * FDIM + ka + 1];
        v2f_t By;  // t_flat[ka, m0+n]
        By.x = t[(size_t)(ka)     * MDIM + m0 + n];
        By.y = t[(size_t)(ka + 1) * MDIM + m0 + n];
        v2f_t Bx;  // t[m0+n, ka]
        Bx.x = t[(size_t)(m0 + n) * FDIM + ka];
        Bx.y = t[(size_t)(m0 + n) * FDIM + ka + 1];
        dy = __builtin_amdgcn_wmma_f32_16x16x4_f32(false, A, false, By, (short)0, dy, false, false);
        dx = __builtin_amdgcn_wmma_f32_16x16x4_f32(false, A, false, Bx, (short)0, dx, false, false);
    }

    int m = m0 + n;
    float scf = cf[m], scg = cg[m], scb = cb[m];
    for (int v = 0; v < 8; ++v) {
        int el = v + 8 * half;                       // D row = e_local
        size_t base = (size_t)m * (4 * EDIM) + wave * 16 + el;
        float yv = dy[v], xv = dx[v];
        out[base + 0 * EDIM] = yv * scf;
        out[base + 1 * EDIM] = xv;
        out[base + 2 * EDIM] = yv * scg;
        out[base + 3 * EDIM] = yv * scb;
    }
}

// ---------------------------------------------------------------------------
// Launch
// ---------------------------------------------------------------------------
extern "C" void kernel_launch(void* const* d_in, const int* in_sizes, int n_in,
                              void* d_out, int out_size, void* d_ws, size_t ws_size,
                              hipStream_t stream) {
    (void)in_sizes; (void)n_in; (void)out_size; (void)ws_size;

    const float* features = (const float*)d_in[0];
    const float* geo_adj  = (const float*)d_in[1];
    const float* fwd_adj  = (const float*)d_in[2];
    const float* bwd_adj  = (const float*)d_in[3];
    const float* geo_nbr  = (const float*)d_in[4];
    const float* fwd_nbr  = (const float*)d_in[5];
    const float* bwd_nbr  = (const float*)d_in[6];
    const float* W_att    = (const float*)d_in[7];
    const float* a1       = (const float*)d_in[8];
    const float* a2       = (const float*)d_in[9];
    const float* weight   = (const float*)d_in[10];
    float* out = (float*)d_out;

    // workspace layout (floats): ~132 KB total; all chunks 16B-aligned
    float* ws  = (float*)d_ws;
    float* w1  = ws;            // 128
    float* w2  = w1 + 128;      // 128
    float* sv  = w2 + 128;      // 4096
    float* uv  = sv + MDIM;     // 4096
    float* rf  = uv + MDIM;     // 4096
    float* rb  = rf + MDIM;     // 4096
    float* rg  = rb + MDIM;     // 4096
    float* cfv = rg + MDIM;     // 4096 (contiguous with cbv, cgv for zeroing)
    float* cbv = cfv + MDIM;    // 4096
    float* cgv = cbv + MDIM;    // 4096

    // 0) zero colsum accumulators
    k_zero<<<(3 * MDIM + 255) / 256, 256, 0, stream>>>(cfv, 3 * MDIM);
    // 1) attention projection vectors
    k_proj_a<<<1, 128, 0, stream>>>(W_att, a1, a2, w1, w2);
    // 2) s = t@w1, u = t@w2
    k_su<<<MDIM / 256, 256, 0, stream>>>(features, w1, w2, sv, uv);
    // 3) row-normalized neighbor averages (r = Pnorm @ u), 3 directions
    k_rowavg<<<dim3(MDIM, 3), 256, 0, stream>>>(fwd_nbr, bwd_nbr, geo_nbr, uv, rf, rb, rg);
    // 4) masked softmax column sums, 3 directions
    k_att<<<dim3(MDIM, 3), 256, 0, stream>>>(fwd_adj, bwd_adj, geo_adj, sv,
                                             rf, rb, rg, cfv, cbv, cgv);
    // 5) WMMA output GEMMs + scaling + concat
    k_out<<<MDIM / 16, 256, 0, stream>>>(features, weight, cfv, cgv, cbv, out);
}